// DynamicsDecoder_74723841016375
// MI455X (gfx1250) — compile-verified
//
#include <hip/hip_runtime.h>
#include <stdint.h>

// MI455X / gfx1250, wave32. All GEMMs via V_WMMA_F32_16X16X4_F32 (exact f32).

typedef __attribute__((ext_vector_type(2))) float v2f;
typedef __attribute__((ext_vector_type(8))) float v8f;

__device__ __forceinline__ v8f wmma4(v2f a, v2f b, v8f c) {
  // 8 args: (neg_a, A, neg_b, B, c_mod, C, reuse_a, reuse_b)
  return __builtin_amdgcn_wmma_f32_16x16x4_f32(false, a, false, b, (short)0, c,
                                               false, false);
}

__device__ __forceinline__ float reluf(float v) { return fmaxf(v, 0.0f); }

static constexpr int B = 128, N = 128, H = 64;
static constexpr int ROWS = B * N; // 16384

// ---------------------------------------------------------------------------
// K0: normalized edge weights  w'[i,j] = keep(i,j)*e[i,j] / max(tw[j],1e-30)
// ---------------------------------------------------------------------------
__global__ __launch_bounds__(128) void k_prep_w(const float* __restrict__ edge,
                                                float* __restrict__ wnorm) {
  int j = threadIdx.x; // one column per thread, N=128 threads
  float tw = 0.0f;
  for (int i = 0; i < N; ++i) {
    float e = edge[i * N + j];
    float wv = (e > 0.01f && i != j) ? e : 0.0f;
    tw += wv;
  }
  float inv = 1.0f / fmaxf(tw, 1e-30f);
  for (int i = 0; i < N; ++i) {
    float e = edge[i * N + j];
    float wv = (e > 0.01f && i != j) ? e : 0.0f;
    wnorm[i * N + j] = wv * inv;
  }
}

// ---------------------------------------------------------------------------
// K1: enc (recomputed from scalar x) -> A = enc@Wm[:H], T = enc@Wm[H:] + bm
// 4 waves/block, each wave owns a 16-row tile of the [16384 x 64] matrix.
// ---------------------------------------------------------------------------
__global__ __launch_bounds__(128) void k_encode_project(
    const float* __restrict__ x, const float* __restrict__ We,
    const float* __restrict__ be, const float* __restrict__ Wm,
    const float* __restrict__ bm, float* __restrict__ Ap,
    float* __restrict__ Tt) {
  __shared__ float sWm[2 * H * H]; // 128x64 = 32 KB
  __shared__ float sWe[H], sBe[H], sBm[H];
  int tid = threadIdx.x;
  for (int t = tid; t < 2 * H * H; t += 128) sWm[t] = Wm[t];
  if (tid < H) { sWe[tid] = We[tid]; sBe[tid] = be[tid]; sBm[tid] = bm[tid]; }
  __syncthreads();

  int wave = tid >> 5;
  int lane = tid & 31;
  int half = lane >> 4;
  int mn = lane & 15;
  int row0 = (blockIdx.x * 4 + wave) * 16;

  float ax = x[row0 + mn]; // scalar input for this lane's A-row

  v8f accP[4], accT[4];
#pragma unroll
  for (int nt = 0; nt < 4; ++nt) { accP[nt] = {}; accT[nt] = {}; }

#pragma unroll
  for (int s = 0; s < 16; ++s) {   // K = 64 in steps of 4
    int c0 = 4 * s + 2 * half;     // shared K base for A and B fragments
    v2f a;
    a.x = reluf(ax * sWe[c0] + sBe[c0]);
    a.y = reluf(ax * sWe[c0 + 1] + sBe[c0 + 1]);
#pragma unroll
    for (int nt = 0; nt < 4; ++nt) {
      int col = nt * 16 + mn;
      v2f bP = {sWm[c0 * H + col], sWm[(c0 + 1) * H + col]};
      v2f bT = {sWm[(H + c0) * H + col], sWm[(H + c0 + 1) * H + col]};
      accP[nt] = wmma4(a, bP, accP[nt]);
      accT[nt] = wmma4(a, bT, accT[nt]);
    }
  }

#pragma unroll
  for (int v = 0; v < 8; ++v) {
    int row = row0 + v + 8 * half;
#pragma unroll
    for (int nt = 0; nt < 4; ++nt) {
      int col = nt * 16 + mn;
      Ap[row * H + col] = accP[nt][v];
      Tt[row * H + col] = accT[nt][v] + sBm[col]; // fold bm into T
    }
  }
}

// ---------------------------------------------------------------------------
// K2: msgs[b,j,h] = sum_i w'[i,j] * relu(A[b,i,h] + T[b,j,h])
// One block per b; A,T tiles in LDS (64 KB); 8 waves x 16 j's; lanes cover h.
// ---------------------------------------------------------------------------
__global__ __launch_bounds__(256) void k_messages(
    const float* __restrict__ Ap, const float* __restrict__ Tt,
    const float* __restrict__ wnorm, float* __restrict__ msgs) {
  __shared__ float sA[N * H]; // 32 KB
  __shared__ float sT[N * H]; // 32 KB
  int b = blockIdx.x;
  int tid = threadIdx.x;
  const float* ApB = Ap + (size_t)b * N * H;
  const float* TtB = Tt + (size_t)b * N * H;
  for (int t = tid; t < N * H; t += 256) { sA[t] = ApB[t]; sT[t] = TtB[t]; }
  __syncthreads();

  int wave = tid >> 5;
  int lane = tid & 31;
  int mn = lane & 15;
  int j0 = wave * 16;

  float t0[16], t1[16], acc0[16], acc1[16];
#pragma unroll
  for (int j = 0; j < 16; ++j) {
    t0[j] = sT[(j0 + j) * H + lane];
    t1[j] = sT[(j0 + j) * H + 32 + lane];
    acc0[j] = 0.0f;
    acc1[j] = 0.0f;
  }

  for (int i = 0; i < N; ++i) {
    float wrow = wnorm[i * N + j0 + mn]; // lane j holds w'[i, j0+j]
    float a0 = sA[i * H + lane];
    float a1 = sA[i * H + 32 + lane];
#pragma unroll
    for (int j = 0; j < 16; ++j) {
      float wv = __shfl(wrow, j, 32);
      acc0[j] = fmaf(wv, reluf(a0 + t0[j]), acc0[j]);
      acc1[j] = fmaf(wv, reluf(a1 + t1[j]), acc1[j]);
    }
  }

  float* mB = msgs + (size_t)b * N * H;
#pragma unroll
  for (int j = 0; j < 16; ++j) {
    mB[(j0 + j) * H + lane] = acc0[j];
    mB[(j0 + j) * H + 32 + lane] = acc1[j];
  }
}

// ---------------------------------------------------------------------------
// K3: hid = relu(msgs@Wp1[:H] + enc@Wp1[H:] + bp1); delta = hid@Wp2 + bp2
//     out = mask ? x : x + delta
// ---------------------------------------------------------------------------
__global__ __launch_bounds__(128) void k_predict(
    const float* __restrict__ x, const float* __restrict__ msgs,
    const float* __restrict__ We, const float* __restrict__ be,
    const float* __restrict__ Wp1, const float* __restrict__ bp1,
    const float* __restrict__ Wp2, const float* __restrict__ bp2,
    const unsigned char* __restrict__ mask, float* __restrict__ out) {
  __shared__ float sW[2 * H * H]; // 32 KB
  __shared__ float sWe[H], sBe[H], sB1[H], sW2[H];
  int tid = threadIdx.x;
  for (int t = tid; t < 2 * H * H; t += 128) sW[t] = Wp1[t];
  if (tid < H) {
    sWe[tid] = We[tid]; sBe[tid] = be[tid];
    sB1[tid] = bp1[tid]; sW2[tid] = Wp2[tid];
  }
  __syncthreads();

  int wave = tid >> 5;
  int lane = tid & 31;
  int half = lane >> 4;
  int mn = lane & 15;
  int row0 = (blockIdx.x * 4 + wave) * 16;

  float ax = x[row0 + mn];

  v8f accU[4], accE[4];
#pragma unroll
  for (int nt = 0; nt < 4; ++nt) { accU[nt] = {}; accE[nt] = {}; }

#pragma unroll
  for (int s = 0; s < 16; ++s) {
    int c0 = 4 * s + 2 * half;
    v2f am = *(const v2f*)(msgs + (size_t)(row0 + mn) * H + c0); // 8B aligned
    v2f ae;
    ae.x = reluf(ax * sWe[c0] + sBe[c0]);
    ae.y = reluf(ax * sWe[c0 + 1] + sBe[c0 + 1]);
#pragma unroll
    for (int nt = 0; nt < 4; ++nt) {
      int col = nt * 16 + mn;
      v2f bU = {sW[c0 * H + col], sW[(c0 + 1) * H + col]};
      v2f bE = {sW[(H + c0) * H + col], sW[(H + c0 + 1) * H + col]};
      accU[nt] = wmma4(am, bU, accU[nt]);
      accE[nt] = wmma4(ae, bE, accE[nt]);
    }
  }

  float bp2v = bp2[0];
  float p[8];
#pragma unroll
  for (int v = 0; v < 8; ++v) {
    float s = 0.0f;
#pragma unroll
    for (int nt = 0; nt < 4; ++nt) {
      int col = nt * 16 + mn;
      float h = reluf(accU[nt][v] + accE[nt][v] + sB1[col]);
      s = fmaf(h, sW2[col], s);
    }
    // reduce over the 16 lanes of this half-wave (cols of the D tile)
    s += __shfl_xor(s, 1, 32);
    s += __shfl_xor(s, 2, 32);
    s += __shfl_xor(s, 4, 32);
    s += __shfl_xor(s, 8, 32);
    p[v] = s;
  }

  if (mn == 0) {
#pragma unroll
    for (int v = 0; v < 8; ++v) {
      int row = row0 + v + 8 * half;
      int n = row & (N - 1);
      float xv = x[row];
      out[row] = mask[n] ? xv : xv + p[v] + bp2v;
    }
  }
}

// ---------------------------------------------------------------------------
extern "C" void kernel_launch(void* const* d_in, const int* in_sizes, int n_in,
                              void* d_out, int out_size, void* d_ws,
                              size_t ws_size, hipStream_t stream) {
  const float* x = (const float*)d_in[0];
  const float* edge = (const float*)d_in[1];
  const unsigned char* mask = (const unsigned char*)d_in[2];
  const float* We = (const float*)d_in[3];
  const float* be = (const float*)d_in[4];
  const float* Wm = (const float*)d_in[5];
  const float* bm = (const float*)d_in[6];
  const float* Wp1 = (const float*)d_in[7];
  const float* bp1 = (const float*)d_in[8];
  const float* Wp2 = (const float*)d_in[9];
  const float* bp2 = (const float*)d_in[10];
  float* out = (float*)d_out;

  const size_t RN = (size_t)ROWS * H; // 1,048,576 floats
  float* ws = (float*)d_ws;
  float* Ap = ws;             // [B*N, H]
  float* Tt = ws + RN;        // [B*N, H]  (bm folded in)
  float* msgs = ws + 2 * RN;  // [B*N, H]
  float* wnorm = ws + 3 * RN; // [N, N]

  k_prep_w<<<1, 128, 0, stream>>>(edge, wnorm);
  k_encode_project<<<ROWS / (16 * 4), 128, 0, stream>>>(x, We, be, Wm, bm, Ap,
                                                        Tt);
  k_messages<<<B, 256, 0, stream>>>(Ap, Tt, wnorm, msgs);
  k_predict<<<ROWS / (16 * 4), 128, 0, stream>>>(x, msgs, We, be, Wp1, bp1,
                                                 Wp2, bp2, mask, out);
}